// Combined_non_max_suppression_32366873542657
// MI455X (gfx1250) — compile-verified
//
#include <hip/hip_runtime.h>
#include <stdint.h>

// ---------------- tunables ----------------
#define NBINS     4096     // score histogram bins over [0,1]
#define CAP       4096     // max candidates kept (power of 2 for bitonic)
#define TARGET    2560     // desired minimum candidate count above cutoff
#define PRE       1024     // candidates whose boxes are pre-gathered to LDS
#define MAX_OUT   256
#define IOU_THR_C   0.5f
#define SCORE_THR_C 0.001f
#define HT        256                 // histo threads per block
#define TILE_FLOATS (HT * 4)          // 1024 floats (4KB) per async tile

// ---------------- gfx1250 async-to-LDS plumbing (guarded) ----------------
#if defined(__HIP_DEVICE_COMPILE__)
  #if __has_builtin(__builtin_amdgcn_global_load_async_to_lds_b128)
    #define HAS_ASYNC 1
  #else
    #define HAS_ASYNC 0
  #endif
  #if __has_builtin(__builtin_amdgcn_s_wait_asynccnt)
    #define WAIT_ASYNC(n) __builtin_amdgcn_s_wait_asynccnt(n)
  #else
    #define WAIT_ASYNC(n) asm volatile("s_wait_asynccnt " #n ::: "memory")
  #endif
#else
  #define HAS_ASYNC 0
  #define WAIT_ASYNC(n)
#endif

#if HAS_ASYNC
typedef int v4i_async __attribute__((vector_size(16)));
typedef __attribute__((address_space(1))) v4i_async* gptr_v4i;
typedef __attribute__((address_space(3))) v4i_async* sptr_v4i;

__device__ __forceinline__ void async_tile_load(const float* g, float* lds_dst) {
  // global_load_async_to_lds_b128: per-lane 16B gather into LDS, tracked by ASYNCcnt.
  __builtin_amdgcn_global_load_async_to_lds_b128(
      (gptr_v4i)(uintptr_t)g,
      (sptr_v4i)(uint32_t)(uintptr_t)lds_dst,
      0, 0);
}
#endif

// ---------------- kernel 0: zero workspace counters ----------------
__global__ void init_kernel(uint32_t* __restrict__ hist) {
  int i = blockIdx.x * blockDim.x + threadIdx.x;
  if (i < NBINS) hist[i] = 0u;
  if (i == 0) hist[NBINS] = 0u;  // candCount
}

// ---------------- kernel 1: score histogram (async double-buffered stream) --
__global__ __launch_bounds__(HT) void histo_kernel(const float* __restrict__ scores,
                                                   int n,
                                                   uint32_t* __restrict__ hist) {
  __shared__ uint32_t lhist[NBINS];
#if HAS_ASYNC
  __shared__ __align__(16) float tile[2][TILE_FLOATS];
#endif
  const int tid = threadIdx.x;
  for (int i = tid; i < NBINS; i += HT) lhist[i] = 0u;
  __syncthreads();

  const int nChunks = n / TILE_FLOATS;

#if HAS_ASYNC
  {
    int c = blockIdx.x;
    if (c < nChunks)
      async_tile_load(scores + (size_t)c * TILE_FLOATS + (size_t)tid * 4,
                      &tile[0][tid * 4]);
    int buf = 0;
    while (c < nChunks) {
      int cn = c + (int)gridDim.x;
      if (cn < nChunks) {
        async_tile_load(scores + (size_t)cn * TILE_FLOATS + (size_t)tid * 4,
                        &tile[buf ^ 1][tid * 4]);
        WAIT_ASYNC(1);   // previous tile complete, next still in flight
      } else {
        WAIT_ASYNC(0);
      }
      asm volatile("" ::: "memory");
      #pragma unroll
      for (int k = 0; k < 4; ++k) {
        float s = tile[buf][tid * 4 + k];
        int b = (int)(s * (float)NBINS);
        b = b < 0 ? 0 : (b > NBINS - 1 ? NBINS - 1 : b);
        atomicAdd(&lhist[b], 1u);
      }
      buf ^= 1;
      c = cn;
    }
  }
#else
  for (int c = blockIdx.x; c < nChunks; c += gridDim.x) {
    const float4* p = (const float4*)(scores + (size_t)c * TILE_FLOATS) + tid;
    __builtin_prefetch(p + gridDim.x * (TILE_FLOATS / 4), 0, 0);
    float4 v = *p;
    float ss[4] = {v.x, v.y, v.z, v.w};
    #pragma unroll
    for (int k = 0; k < 4; ++k) {
      int b = (int)(ss[k] * (float)NBINS);
      b = b < 0 ? 0 : (b > NBINS - 1 ? NBINS - 1 : b);
      atomicAdd(&lhist[b], 1u);
    }
  }
#endif

  // tail elements (n not multiple of TILE_FLOATS)
  if (blockIdx.x == 0) {
    for (int i = nChunks * TILE_FLOATS + tid; i < n; i += HT) {
      float s = scores[i];
      int b = (int)(s * (float)NBINS);
      b = b < 0 ? 0 : (b > NBINS - 1 ? NBINS - 1 : b);
      atomicAdd(&lhist[b], 1u);
    }
  }

  __syncthreads();
  for (int i = tid; i < NBINS; i += HT) {
    uint32_t v = lhist[i];
    if (v) atomicAdd(&hist[i], v);
  }
}

// ---------------- kernel 2: find cutoff bin (suffix-sum from top) ----------
__global__ void cutoff_kernel(const uint32_t* __restrict__ hist,
                              int* __restrict__ cutoff) {
  if (blockIdx.x == 0 && threadIdx.x == 0) {
    uint32_t acc = 0;
    int c = 0;
    for (int b = NBINS - 1; b >= 0; --b) {
      acc += hist[b];
      if (acc >= TARGET) { c = b; break; }
    }
    *cutoff = c;
  }
}

// ---------------- kernel 3: compact candidates above cutoff ----------------
__global__ __launch_bounds__(256) void compact_kernel(const float* __restrict__ scores,
                                                      int n,
                                                      const int* __restrict__ cutoff,
                                                      uint32_t* __restrict__ candCount,
                                                      uint64_t* __restrict__ cand) {
  const int c = *cutoff;
  const int n4 = n >> 2;
  const float4* s4 = (const float4*)scores;
  const int stride = gridDim.x * blockDim.x;
  for (int i = blockIdx.x * blockDim.x + threadIdx.x; i < n4; i += stride) {
    __builtin_prefetch(s4 + i + stride, 0, 0);
    float4 v = s4[i];
    float ss[4] = {v.x, v.y, v.z, v.w};
    #pragma unroll
    for (int k = 0; k < 4; ++k) {
      float s = ss[k];
      int b = (int)(s * (float)NBINS);
      if (b >= c && s > SCORE_THR_C) {
        uint32_t pos = atomicAdd(candCount, 1u);
        if (pos < CAP) {
          uint32_t idx = (uint32_t)(i * 4 + k);
          // key: sortable score bits (positive floats) | ~idx so ties pick lowest idx
          cand[pos] = ((uint64_t)__float_as_uint(s) << 32) | (uint32_t)(~idx);
        }
      }
    }
  }
  if (blockIdx.x == 0 && threadIdx.x == 0) {   // scalar tail
    for (int i = n4 * 4; i < n; ++i) {
      float s = scores[i];
      int b = (int)(s * (float)NBINS);
      if (b >= c && s > SCORE_THR_C) {
        uint32_t pos = atomicAdd(candCount, 1u);
        if (pos < CAP)
          cand[pos] = ((uint64_t)__float_as_uint(s) << 32) | (uint32_t)(~(uint32_t)i);
      }
    }
  }
}

// ---------------- kernel 4: bitonic sort + greedy NMS + output -------------
__global__ __launch_bounds__(1024) void sort_greedy_kernel(const float* __restrict__ boxes,
                                                           const uint64_t* __restrict__ cand,
                                                           const uint32_t* __restrict__ candCount,
                                                           int* __restrict__ outIdx,
                                                           float* __restrict__ outSc) {
  __shared__ uint64_t keys[CAP];                       // 32 KB
  __shared__ float cy1[PRE], cx1[PRE], cy2[PRE], cx2[PRE], car[PRE];  // 20 KB
  __shared__ float sy1[MAX_OUT], sx1[MAX_OUT], sy2[MAX_OUT], sx2[MAX_OUT], sar[MAX_OUT]; // 5 KB

  const int tid = threadIdx.x;
  const int cnt0 = (int)*candCount;
  const int cnt = cnt0 < CAP ? cnt0 : CAP;

  for (int i = tid; i < CAP; i += blockDim.x) keys[i] = (i < cnt) ? cand[i] : 0ull;
  for (int i = tid; i < MAX_OUT; i += blockDim.x) { outIdx[i] = -1; outSc[i] = 0.0f; }
  __syncthreads();

  // bitonic sort, descending by key
  for (int k = 2; k <= CAP; k <<= 1) {
    for (int j = k >> 1; j > 0; j >>= 1) {
      for (int i = tid; i < CAP; i += blockDim.x) {
        int ixj = i ^ j;
        if (ixj > i) {
          uint64_t a = keys[i], b = keys[ixj];
          bool up = ((i & k) == 0);
          bool sw = up ? (a < b) : (a > b);   // descending overall
          if (sw) { keys[i] = b; keys[ixj] = a; }
        }
      }
      __syncthreads();
    }
  }

  // pre-gather + canonicalize boxes of top PRE candidates (coalesced-ish gather)
  for (int i = tid; i < PRE; i += blockDim.x) {
    if (i < cnt) {
      uint64_t key = keys[i];
      int idx = (int)~((uint32_t)key);
      float b0 = boxes[(size_t)idx * 4 + 0];
      float b1 = boxes[(size_t)idx * 4 + 1];
      float b2 = boxes[(size_t)idx * 4 + 2];
      float b3 = boxes[(size_t)idx * 4 + 3];
      float y1 = fminf(b0, b2), x1 = fminf(b1, b3);
      float y2 = fmaxf(b0, b2), x2 = fmaxf(b1, b3);
      cy1[i] = y1; cx1[i] = x1; cy2[i] = y2; cx2[i] = x2;
      car[i] = (y2 - y1) * (x2 - x1);
    }
  }
  __syncthreads();

  // greedy accept on one wave32: lanes split the IoU-vs-selected checks
  if (tid < 32) {
    const int lane = tid;
    int sel = 0;
    for (int i = 0; i < CAP && sel < MAX_OUT; ++i) {
      uint64_t key = keys[i];
      float sc = __uint_as_float((uint32_t)(key >> 32));
      if (!(sc > SCORE_THR_C)) break;        // padding (key=0) also stops here
      float y1, x1, y2, x2, ar;
      if (i < PRE) {
        y1 = cy1[i]; x1 = cx1[i]; y2 = cy2[i]; x2 = cx2[i]; ar = car[i];
      } else {
        int idx = (int)~((uint32_t)key);
        float b0 = boxes[(size_t)idx * 4 + 0];
        float b1 = boxes[(size_t)idx * 4 + 1];
        float b2 = boxes[(size_t)idx * 4 + 2];
        float b3 = boxes[(size_t)idx * 4 + 3];
        y1 = fminf(b0, b2); x1 = fminf(b1, b3);
        y2 = fmaxf(b0, b2); x2 = fmaxf(b1, b3);
        ar = (y2 - y1) * (x2 - x1);
      }
      int supp = 0;
      for (int j = lane; j < sel; j += 32) {
        float ih = fmaxf(0.0f, fminf(y2, sy2[j]) - fmaxf(y1, sy1[j]));
        float iw = fmaxf(0.0f, fminf(x2, sx2[j]) - fmaxf(x1, sx1[j]));
        float inter = ih * iw;
        float uni = ar + sar[j] - inter;
        float iou = (uni > 0.0f) ? (inter / uni) : 0.0f;
        if (iou > IOU_THR_C) { supp = 1; break; }
      }
      if (!__any(supp)) {
        if (lane == 0) {
          sy1[sel] = y1; sx1[sel] = x1; sy2[sel] = y2; sx2[sel] = x2; sar[sel] = ar;
          outIdx[sel] = (int)~((uint32_t)key);
          outSc[sel] = sc;
        }
        ++sel;
      }
    }
  }
}

// ---------------- launcher ----------------
extern "C" void kernel_launch(void* const* d_in, const int* in_sizes, int n_in,
                              void* d_out, int out_size, void* d_ws, size_t ws_size,
                              hipStream_t stream) {
  const float* boxes  = (const float*)d_in[0];
  const float* scores = (const float*)d_in[1];
  const int n = in_sizes[1];   // pred_conf flat count == N

  // ws layout: [0..NBINS-1] hist | [NBINS] candCount | [NBINS+1] cutoff |
  //            [NBINS+8 ..] uint64 cand[CAP]  (byte offset 16416, 8B aligned)
  uint32_t* hist      = (uint32_t*)d_ws;
  uint32_t* candCount = hist + NBINS;
  int*      cutoff    = (int*)(hist + NBINS + 1);
  uint64_t* cand      = (uint64_t*)(hist + NBINS + 8);

  int* outIdx  = (int*)d_out;
  float* outSc = (float*)d_out + MAX_OUT;

  init_kernel<<<(NBINS + 255) / 256, 256, 0, stream>>>(hist);
  histo_kernel<<<512, HT, 0, stream>>>(scores, n, hist);
  cutoff_kernel<<<1, 64, 0, stream>>>(hist, cutoff);
  compact_kernel<<<1024, 256, 0, stream>>>(scores, n, cutoff, candCount, cand);
  sort_greedy_kernel<<<1, 1024, 0, stream>>>(boxes, cand, candCount, outIdx, outSc);
}